// OmegaEntangle_40407052320831
// MI455X (gfx1250) — compile-verified
//
#include <hip/hip_runtime.h>

typedef __attribute__((ext_vector_type(2))) float v2f;
typedef __attribute__((ext_vector_type(4))) float v4f;
typedef __attribute__((ext_vector_type(8))) float v8f;

#define NDOM 512
#define VDIM 32768
#define SDIM 2048

#define WMMA_F32(A, B, C) \
    __builtin_amdgcn_wmma_f32_16x16x4_f32(false, (A), false, (B), (short)0, (C), false, false)

// ---------------------------------------------------------------------------
// Kernel 1: per-domain reduction  p[d] = sum(v^2), m[d] = mean(v)
// 64MB stream (the HBM-bandwidth-dominant stage) -> non-temporal b128 loads.
// ---------------------------------------------------------------------------
__global__ void __launch_bounds__(256)
k_reduce(const float* __restrict__ vulns, float* __restrict__ pm)
{
    const int row = blockIdx.x;
    const v4f* base = (const v4f*)(vulns + (size_t)row * VDIM);
    float s = 0.0f, sq = 0.0f;
    for (int i = threadIdx.x; i < VDIM / 4; i += 256) {
        v4f x = __builtin_nontemporal_load(&base[i]);
        s  += (x.x + x.y) + (x.z + x.w);
        sq += (x.x * x.x + x.y * x.y) + (x.z * x.z + x.w * x.w);
    }
    __shared__ float ss[256];
    __shared__ float ssq[256];
    ss[threadIdx.x] = s;
    ssq[threadIdx.x] = sq;
    __syncthreads();
    for (int off = 128; off > 0; off >>= 1) {
        if (threadIdx.x < off) {
            ss[threadIdx.x]  += ss[threadIdx.x + off];
            ssq[threadIdx.x] += ssq[threadIdx.x + off];
        }
        __syncthreads();
    }
    if (threadIdx.x == 0) {
        pm[row]        = ssq[0];                    // p
        pm[NDOM + row] = ss[0] * (1.0f / VDIM);     // m
    }
}

// ---------------------------------------------------------------------------
// Kernel 2: materialize C_re/C_im [512x512] once (strict upper mask).
// exp(i*atan2(m_j,m_i)) = (m_i + i*m_j)/sqrt(m_i^2+m_j^2) -> no trig.
// ---------------------------------------------------------------------------
__global__ void __launch_bounds__(256)
k_coeff(const float* __restrict__ pm, float* __restrict__ Cre, float* __restrict__ Cim)
{
    const int idx = blockIdx.x * 256 + threadIdx.x;   // 0..262143
    const int i = idx >> 9;
    const int j = idx & (NDOM - 1);
    const float p_i = pm[i], m_i = pm[NDOM + i];
    const float p_j = pm[j], m_j = pm[NDOM + j];
    const float amp = __builtin_sqrtf(p_i * p_j) *
                      __builtin_amdgcn_rsqf(m_i * m_i + m_j * m_j);
    const bool u = i < j;
    Cre[idx] = u ? amp * m_i : 0.0f;
    Cim[idx] = u ? amp * m_j : 0.0f;
}

// ---------------------------------------------------------------------------
// Kernel 3: Egt[m][k] = embed[ids[k]][m]  (gather + transpose, k-major).
// LDS 32x33 tile transpose keeps both sides coalesced; folds the ids[]
// indirection out of the GEMM inner loops.
// ---------------------------------------------------------------------------
__global__ void __launch_bounds__(256)
k_gather_t(const float* __restrict__ embed, const int* __restrict__ ids,
           float* __restrict__ Egt)
{
    __shared__ float tile[32][33];
    const int kt = blockIdx.x & 15;    // 16 k-tiles   (512/32)
    const int mt = blockIdx.x >> 4;    // 64 m-tiles   (2048/32)
    const int tx = threadIdx.x & 31;
    const int ty = threadIdx.x >> 5;   // 0..7
    #pragma unroll
    for (int s = 0; s < 4; ++s) {
        const int k = kt * 32 + ty + 8 * s;
        tile[ty + 8 * s][tx] = embed[(size_t)ids[k] * SDIM + mt * 32 + tx];
    }
    __syncthreads();
    #pragma unroll
    for (int s = 0; s < 4; ++s) {
        const int m = mt * 32 + ty + 8 * s;
        Egt[(size_t)m * NDOM + kt * 32 + tx] = tile[tx][ty + 8 * s];
    }
}

// D-fragment: VGPR r holds element (M = mbase + r + 8*half, N = lane%16).
// 8 consecutive M per lane -> contiguous b128 stores into k-major Tt[n][m].
__device__ __forceinline__ void store_col8(float* dst, const v8f a)
{
    v4f lo = {a[0], a[1], a[2], a[3]};
    v4f hi = {a[4], a[5], a[6], a[7]};
    *(v4f*)dst       = lo;
    *(v4f*)(dst + 4) = hi;
}

// ---------------------------------------------------------------------------
// Kernel 4: Tt = (C * E)^T  for re & im.  M=512,K=512,N=2048.
// 32x32 output per wave (2x2 tiles x 2 components = 8 v8f accumulators).
// All fragment loads are contiguous v2f (kA,kA+1 pairs per half-wave).
// 6 b64 loads : 8 WMMAs per k-step.
// ---------------------------------------------------------------------------
__global__ void __launch_bounds__(256)
k_gemm1(const float* __restrict__ Cre, const float* __restrict__ Cim,
        const float* __restrict__ Egt,
        float* __restrict__ Ttre, float* __restrict__ Ttim)
{
    const int wave = (blockIdx.x * 256 + threadIdx.x) >> 5;  // 0..1023
    const int lane = threadIdx.x & 31;
    const int tm   = wave >> 6;        // 0..15  (32-row strips of C)
    const int tn   = wave & 63;        // 0..63  (32-col strips of E)
    const int half = lane >> 4;
    const int l16  = lane & 15;

    const int m0 = tm * 32 + l16;
    const int n0 = tn * 32 + l16;

    const v2f* Ar0 = (const v2f*)(Cre + (size_t)m0 * NDOM);
    const v2f* Ar1 = (const v2f*)(Cre + (size_t)(m0 + 16) * NDOM);
    const v2f* Ai0 = (const v2f*)(Cim + (size_t)m0 * NDOM);
    const v2f* Ai1 = (const v2f*)(Cim + (size_t)(m0 + 16) * NDOM);
    const v2f* B0  = (const v2f*)(Egt + (size_t)n0 * NDOM);
    const v2f* B1  = (const v2f*)(Egt + (size_t)(n0 + 16) * NDOM);

    v8f r00 = {}, r01 = {}, r10 = {}, r11 = {};
    v8f i00 = {}, i01 = {}, i10 = {}, i11 = {};

    for (int k4 = 0; k4 < NDOM; k4 += 4) {
        const int kh = (k4 >> 1) + half;     // v2f index of pair (kA, kA+1)
        const v2f ar0 = Ar0[kh], ar1 = Ar1[kh];
        const v2f ai0 = Ai0[kh], ai1 = Ai1[kh];
        const v2f b0  = B0[kh],  b1  = B1[kh];
        r00 = WMMA_F32(ar0, b0, r00);
        r01 = WMMA_F32(ar0, b1, r01);
        r10 = WMMA_F32(ar1, b0, r10);
        r11 = WMMA_F32(ar1, b1, r11);
        i00 = WMMA_F32(ai0, b0, i00);
        i01 = WMMA_F32(ai0, b1, i01);
        i10 = WMMA_F32(ai1, b0, i10);
        i11 = WMMA_F32(ai1, b1, i11);
    }

    const int mb0 = tm * 32 + 8 * half;
    const int mb1 = mb0 + 16;
    store_col8(Ttre + (size_t)n0 * NDOM + mb0,        r00);
    store_col8(Ttre + (size_t)(n0 + 16) * NDOM + mb0, r01);
    store_col8(Ttre + (size_t)n0 * NDOM + mb1,        r10);
    store_col8(Ttre + (size_t)(n0 + 16) * NDOM + mb1, r11);
    store_col8(Ttim + (size_t)n0 * NDOM + mb0,        i00);
    store_col8(Ttim + (size_t)(n0 + 16) * NDOM + mb0, i01);
    store_col8(Ttim + (size_t)n0 * NDOM + mb1,        i10);
    store_col8(Ttim + (size_t)(n0 + 16) * NDOM + mb1, i11);
}

// ---------------------------------------------------------------------------
// Kernel 5: out = E^T * T  (re & im), interleaved complex64 output.
// M=N=2048, K=512.  Same 2x2-tile wave structure; A comes from Egt (shared by
// both components), B from k-major Tt -> 6 b64 loads : 8 WMMAs per k-step.
// Output stores are non-temporal (32MB, never re-read).
// ---------------------------------------------------------------------------
__global__ void __launch_bounds__(256)
k_gemm2(const float* __restrict__ Egt, const float* __restrict__ Ttre,
        const float* __restrict__ Ttim, float* __restrict__ out)
{
    const int wave = (blockIdx.x * 256 + threadIdx.x) >> 5;  // 0..4095
    const int lane = threadIdx.x & 31;
    const int tm   = wave >> 6;        // 0..63
    const int tn   = wave & 63;        // 0..63
    const int half = lane >> 4;
    const int l16  = lane & 15;

    const int m0 = tm * 32 + l16;
    const int n0 = tn * 32 + l16;

    const v2f* A0  = (const v2f*)(Egt  + (size_t)m0 * NDOM);
    const v2f* A1  = (const v2f*)(Egt  + (size_t)(m0 + 16) * NDOM);
    const v2f* Br0 = (const v2f*)(Ttre + (size_t)n0 * NDOM);
    const v2f* Br1 = (const v2f*)(Ttre + (size_t)(n0 + 16) * NDOM);
    const v2f* Bi0 = (const v2f*)(Ttim + (size_t)n0 * NDOM);
    const v2f* Bi1 = (const v2f*)(Ttim + (size_t)(n0 + 16) * NDOM);

    v8f r00 = {}, r01 = {}, r10 = {}, r11 = {};
    v8f i00 = {}, i01 = {}, i10 = {}, i11 = {};

    for (int k4 = 0; k4 < NDOM; k4 += 4) {
        const int kh = (k4 >> 1) + half;
        const v2f a0  = A0[kh],  a1  = A1[kh];
        const v2f br0 = Br0[kh], br1 = Br1[kh];
        const v2f bi0 = Bi0[kh], bi1 = Bi1[kh];
        r00 = WMMA_F32(a0, br0, r00);
        r01 = WMMA_F32(a0, br1, r01);
        r10 = WMMA_F32(a1, br0, r10);
        r11 = WMMA_F32(a1, br1, r11);
        i00 = WMMA_F32(a0, bi0, i00);
        i01 = WMMA_F32(a0, bi1, i01);
        i10 = WMMA_F32(a1, bi0, i10);
        i11 = WMMA_F32(a1, bi1, i11);
    }

    v2f* outc = (v2f*)out;   // interleaved complex
    #pragma unroll
    for (int mi = 0; mi < 2; ++mi) {
        const v8f* ar = mi ? &r10 : &r00;
        const v8f* ai = mi ? &i10 : &i00;
        #pragma unroll
        for (int ni = 0; ni < 2; ++ni) {
            const v8f re = ni ? (mi ? r11 : r01) : (mi ? r10 : r00);
            const v8f im = ni ? (mi ? i11 : i01) : (mi ? i10 : i00);
            const int nc = tn * 32 + ni * 16 + l16;
            const int mbase = tm * 32 + mi * 16 + 8 * half;
            #pragma unroll
            for (int r = 0; r < 8; ++r) {
                v2f c = {re[r], im[r]};
                __builtin_nontemporal_store(c, &outc[(size_t)(mbase + r) * SDIM + nc]);
            }
        }
        (void)ar; (void)ai;
    }
}

// ---------------------------------------------------------------------------
extern "C" void kernel_launch(void* const* d_in, const int* in_sizes, int n_in,
                              void* d_out, int out_size, void* d_ws, size_t ws_size,
                              hipStream_t stream)
{
    const float* vulns = (const float*)d_in[0];   // [512, 32768] f32
    const float* embed = (const float*)d_in[1];   // [512, 2048]  f32
    const int*   ids   = (const int*)d_in[2];     // [512] i32
    float*       out   = (float*)d_out;           // [2048,2048] complex64 interleaved

    // workspace layout (floats): Egt 1M | Ttre 1M | Ttim 1M | Cre 256K | Cim 256K | pm 1K
    float* Egt  = (float*)d_ws;
    float* Ttre = Egt  + (size_t)SDIM * NDOM;
    float* Ttim = Ttre + (size_t)SDIM * NDOM;
    float* Cre  = Ttim + (size_t)SDIM * NDOM;
    float* Cim  = Cre  + (size_t)NDOM * NDOM;
    float* pm   = Cim  + (size_t)NDOM * NDOM;

    k_reduce  <<<NDOM, 256, 0, stream>>>(vulns, pm);
    k_coeff   <<<(NDOM * NDOM) / 256, 256, 0, stream>>>(pm, Cre, Cim);
    k_gather_t<<<(NDOM / 32) * (SDIM / 32), 256, 0, stream>>>(embed, ids, Egt);
    k_gemm1   <<<(NDOM / 32) * (SDIM / 32) / 8, 256, 0, stream>>>(Cre, Cim, Egt, Ttre, Ttim);
    k_gemm2   <<<(SDIM / 32) * (SDIM / 32) / 8, 256, 0, stream>>>(Egt, Ttre, Ttim, out);
}